// MorletTransform_9191230014010
// MI455X (gfx1250) — compile-verified
//
#include <hip/hip_runtime.h>
#include <math.h>

typedef __attribute__((ext_vector_type(2))) float v2f;
typedef __attribute__((ext_vector_type(4))) float v4f;
typedef __attribute__((ext_vector_type(8))) float v8f;

#define WIN        1024
#define NHARM      64
#define SRATE      16000.0f
#define TP         16000.0f          // sample_rate / half_bandwidth
#define TWO_PI_F   6.28318530717958647692f

// ds_swizzle_b32 group-of-32 encodings: offset = xor_mask<<10 | and_mask(0x1F)
#define SWZ_XOR1   0x041F
#define SWZ_XOR2   0x081F
#define SWZ_XOR4   0x101F
#define SWZ_XOR8   0x201F
#define SWZ_XOR16  0x401F
#define SWZ_ADD(x, imm) \
  ((x) + __int_as_float(__builtin_amdgcn_ds_swizzle(__float_as_int(x), (imm))))

// One frame per workgroup; 4 waves, wave w handles harmonics 16w+1 .. 16w+16.
// Inner contraction via V_WMMA_F32_16X16X4_F32 using the phase factorization
//   e^{-i th k n} = e^{-i th k i} * e^{-i th k (n0+4j)},  n = n0 + 4j + i.
__global__ __launch_bounds__(128) void morlet_wmma_kernel(
    const float* __restrict__ audio,   // (F, 1024)
    const float* __restrict__ f0,      // (F,)
    float* __restrict__ out,           // hd (F*64) then amp (F)
    int n_frames)
{
  __shared__ float sa[WIN];      // normalizer * gauss * audio
  __shared__ float smag[NHARM];  // |transform|
  __shared__ float spart[4];     // per-wave amplitude partials

  const int frame = blockIdx.x;
  const int tid   = threadIdx.x;
  const float* af = audio + (size_t)frame * WIN;

  // ---- issue the global loads FIRST (2x global_load_b128 per thread);
  // their latency is hidden under the sincos setup below.
  v4f w0 = *(const v4f*)(af + tid * 8);
  v4f w1 = *(const v4f*)(af + tid * 8 + 4);

  const float f0v   = f0[frame];
  const float theta = TWO_PI_F * f0v / SRATE;
  const float norm  = 1.0f / sqrtf(3.14159265358979323846f * TP);

  const int lane = tid & 31;
  const int wave = tid >> 5;
  const int hi   = lane >> 4;   // half-wave select
  const int j    = lane & 15;   // N column / sample-chunk index

  // ---- A operand: A_re[m,k]=cos(th*h*k), A_im[m,k]=-sin(th*h*k), k in {0..3}
  // Lane layout (ISA 7.12.2, 32-bit A 16x4): M = lane&15;
  // lanes 0-15 hold K={0,1}, lanes 16-31 hold K={2,3}.
  v2f a_re, a_im;
  {
    int   mA  = lane & 15;
    float psi = theta * (float)(16 * wave + mA + 1);
    float s1, c1;
    sincosf(psi, &s1, &c1);
    float c2 = c1 * c1 - s1 * s1;
    float s2 = 2.0f * c1 * s1;
    float c3 = c1 * c2 - s1 * s2;
    float s3 = s1 * c2 + c1 * s2;
    a_re.x = hi ?  c2 : 1.0f;
    a_re.y = hi ?  c3 : c1;
    a_im.x = hi ? -s2 : 0.0f;
    a_im.y = hi ? -s3 : -s1;
  }

  // ---- per-row twiddle state: e^{-i th h (n0+4j)}, advanced 64 samples/block
  float tc[8], ts[8], rc[8], rs[8], accr[8], acci[8];
  #pragma unroll
  for (int r = 0; r < 8; ++r) {
    int   h   = 16 * wave + r + 8 * hi + 1;   // harmonic number (1-based)
    float ph0 = theta * (float)h * (float)(4 * j);
    float dph = theta * (float)h * 64.0f;
    sincosf(ph0, &ts[r], &tc[r]);
    sincosf(dph, &rs[r], &rc[r]);
    accr[r] = 0.0f;
    acci[r] = 0.0f;
  }

  // ---- window + stage audio into LDS (loads have long since landed)
  {
    float w[8] = {w0.x, w0.y, w0.z, w0.w, w1.x, w1.y, w1.z, w1.w};
    #pragma unroll
    for (int i = 0; i < 8; ++i) {
      int   n = tid * 8 + i;
      float d = (float)(n - WIN / 2);
      sa[n] = norm * __expf(-d * d * (1.0f / TP)) * w[i];
    }
  }
  __syncthreads();

  const v8f czero = (v8f)0.0f;

  // ---- main loop: 16 blocks of 64 samples, 2 WMMAs each (Re/Im)
  for (int blk = 0; blk < 16; ++blk) {
    // B operand (4x16): rows {0,1} in lanes 0-15, rows {2,3} in lanes 16-31.
    // float2 at sa[n0 + 4j + 2*hi] -> 8-byte aligned ds_load_b64.
    int base = blk * 64 + 4 * j + 2 * hi;
    v2f b = *(const v2f*)&sa[base];

    v8f pre = __builtin_amdgcn_wmma_f32_16x16x4_f32(
        false, a_re, false, b, (short)0, czero, false, false);
    v8f pim = __builtin_amdgcn_wmma_f32_16x16x4_f32(
        false, a_im, false, b, (short)0, czero, false, false);

    #pragma unroll
    for (int r = 0; r < 8; ++r) {
      float Pr = pre[r], Pi = pim[r];
      // acc += e^{-i phi} * P  with (tc,ts) = (cos phi, sin phi)
      accr[r] = fmaf(tc[r], Pr, fmaf( ts[r], Pi, accr[r]));
      acci[r] = fmaf(tc[r], Pi, fmaf(-ts[r], Pr, acci[r]));
      // rotate twiddle by d_phi = th*h*64
      float nc = tc[r] * rc[r] - ts[r] * rs[r];
      float ns = ts[r] * rc[r] + tc[r] * rs[r];
      tc[r] = nc; ts[r] = ns;
    }
  }

  // ---- butterfly over j within each 16-lane half: one ds_swizzle per step
  #pragma unroll
  for (int r = 0; r < 8; ++r) {
    accr[r] = SWZ_ADD(accr[r], SWZ_XOR1);
    acci[r] = SWZ_ADD(acci[r], SWZ_XOR1);
    accr[r] = SWZ_ADD(accr[r], SWZ_XOR2);
    acci[r] = SWZ_ADD(acci[r], SWZ_XOR2);
    accr[r] = SWZ_ADD(accr[r], SWZ_XOR4);
    acci[r] = SWZ_ADD(acci[r], SWZ_XOR4);
    accr[r] = SWZ_ADD(accr[r], SWZ_XOR8);
    acci[r] = SWZ_ADD(acci[r], SWZ_XOR8);
  }

  // ---- every lane now holds its half's reduced values; magnitudes + mask
  float mag[8];
  float s8 = 0.0f;
  #pragma unroll
  for (int r = 0; r < 8; ++r) {
    int   hidx = 16 * wave + r + 8 * hi;            // 0-based harmonic
    float fc   = f0v * (float)(hidx + 1) / SRATE;
    float m    = sqrtf(accr[r] * accr[r] + acci[r] * acci[r]);
    mag[r] = (fc > 0.5f) ? 0.0f : m;
    s8 += mag[r];
  }
  // combine the two 16-lane halves -> per-wave 16-harmonic sum (all lanes)
  float s16 = SWZ_ADD(s8, SWZ_XOR16);

  if (lane == 0) spart[wave] = s16;
  if (j == 0) {
    #pragma unroll
    for (int r = 0; r < 8; ++r) smag[16 * wave + r + 8 * hi] = mag[r];
  }
  __syncthreads();

  const float amp = spart[0] + spart[1] + spart[2] + spart[3];
  if (tid < NHARM) {
    out[(size_t)frame * NHARM + tid] = smag[tid] / amp;
  }
  if (tid == 0) {
    float ac = fminf(fmaxf(amp * 2.0f, 0.0f), 1.0f);
    out[(size_t)n_frames * NHARM + frame] = ac;
  }
}

extern "C" void kernel_launch(void* const* d_in, const int* in_sizes, int n_in,
                              void* d_out, int out_size, void* d_ws, size_t ws_size,
                              hipStream_t stream) {
  const float* audio = (const float*)d_in[0];
  const float* f0    = (const float*)d_in[1];
  int n_frames = in_sizes[1];   // 2*250*1 = 500 frames
  morlet_wmma_kernel<<<dim3(n_frames), dim3(128), 0, stream>>>(
      audio, f0, (float*)d_out, n_frames);
}